// HardTripletMiningLoss_76579266887991
// MI455X (gfx1250) — compile-verified
//
#include <hip/hip_runtime.h>

// ---------------------------------------------------------------------------
// HardTripletMiningLoss on gfx1250 (CDNA5, wave32)
//   N = 480 embeddings, D = 128, margin A = 0.2
//   Gram via V_WMMA_F32_16X16X4_F32 (f32 precision preserved: the td>0
//   threshold is precision-sensitive, so no bf16/fp8 downcast).
//   Triplet pass factored per-row: O(N^2 * |P_i|) instead of O(N^3).
// ---------------------------------------------------------------------------

typedef __attribute__((ext_vector_type(2))) float v2f;
typedef __attribute__((ext_vector_type(8))) float v8f;

#define N_EMB   480
#define DIM     128
#define MARGIN  0.2f
#define TILES   (N_EMB / 16)   // 30

// ---- workspace layout (in floats) ----
// [0            , 61440)  : emb  (480*128)
// [61440        , 61920)  : sq   (480)
// [61920        , 62400)  : per-row partial sums (480)
// [62400        , 62880)  : per-row partial counts (480, as uint bits)
// [62880        , 293280) : pd   (480*480)
#define WS_EMB   0
#define WS_SQ    61440
#define WS_SUM   61920
#define WS_CNT   62400
#define WS_PD    62880

// ---------------------------------------------------------------------------
__global__ void concat_kernel(const float* __restrict__ a,
                              const float* __restrict__ p,
                              const float* __restrict__ n,
                              float* __restrict__ emb) {
    int idx = blockIdx.x * blockDim.x + threadIdx.x;
    if (idx >= N_EMB * DIM) return;
    const int b1 = 160 * DIM, b2 = 320 * DIM;
    float v;
    if (idx < b1)      v = a[idx];
    else if (idx < b2) v = p[idx - b1];
    else               v = n[idx - b2];
    emb[idx] = v;
}

// ---------------------------------------------------------------------------
// one wave per row: sq[row] = sum(emb[row,:]^2)
__global__ void sq_kernel(const float* __restrict__ emb,
                          float* __restrict__ sq) {
    int row  = blockIdx.x;
    int lane = threadIdx.x;                  // 0..31 (wave32)
    const float* p = emb + row * DIM;
    float s = 0.0f;
    #pragma unroll
    for (int k = lane; k < DIM; k += 32) { float v = p[k]; s += v * v; }
    #pragma unroll
    for (int off = 16; off > 0; off >>= 1) s += __shfl_down(s, off, 32);
    if (lane == 0) sq[row] = s;
}

// ---------------------------------------------------------------------------
// One wave computes one 16x16 tile of pd = relu(sq_i + sq_j - 2 * emb@emb^T)
// using V_WMMA_F32_16X16X4_F32, K-loop over DIM in steps of 4.
//
// A-matrix (16x4 f32, ISA 7.12.2): lanes 0-15 hold M=lane, VGPR0=K0,VGPR1=K1;
// lanes 16-31 hold M=lane-16, VGPR0=K2,VGPR1=K3  -> contiguous float2 load.
// B-matrix (4x16) mirrors with N striped across lanes.
// C/D: VGPR r holds M = r + 8*(lane>=16), N = lane&15.
__global__ void pd_wmma_kernel(const float* __restrict__ emb,
                               const float* __restrict__ sq,
                               float* __restrict__ pd) {
    const int tj   = blockIdx.x;             // column tile
    const int ti   = blockIdx.y;             // row tile
    const int lane = threadIdx.x;            // 0..31
    const int lo   = lane & 15;
    const int hi   = lane >> 4;

    const float* __restrict__ arow = emb + (ti * 16 + lo) * DIM + 2 * hi;
    const float* __restrict__ brow = emb + (tj * 16 + lo) * DIM + 2 * hi;

    v8f c = {};
    #pragma unroll 4
    for (int k = 0; k < DIM; k += 4) {
        v2f a, b;
        a.x = arow[k]; a.y = arow[k + 1];
        b.x = brow[k]; b.y = brow[k + 1];
        // 8 args: (neg_a, A, neg_b, B, c_mod, C, reuse_a, reuse_b)
        c = __builtin_amdgcn_wmma_f32_16x16x4_f32(
                false, a, false, b, (short)0, c, false, false);
    }

    const int   col = tj * 16 + lo;
    const float sqc = sq[col];
    #pragma unroll
    for (int r = 0; r < 8; ++r) {
        int   m = ti * 16 + r + 8 * hi;
        float v = sq[m] + sqc - 2.0f * c[r];
        pd[m * N_EMB + col] = v > 0.0f ? v : 0.0f;
    }
}

// ---------------------------------------------------------------------------
// One 256-thread block per anchor row i. Deterministic: fixed thread->k map,
// fixed j order, per-block result written to a private slot (no atomics).
__global__ void triplet_reduce_kernel(const float* __restrict__ pd,
                                      const int* __restrict__ ind,
                                      float* __restrict__ part_sum,
                                      unsigned int* __restrict__ part_cnt) {
    __shared__ float    s_pd[N_EMB];
    __shared__ int      s_lab[N_EMB];
    __shared__ int      s_pos[N_EMB];
    __shared__ int      s_npos;
    __shared__ float    red_s[8];
    __shared__ unsigned red_c[8];

    const int i = blockIdx.x;
    const int t = threadIdx.x;               // 0..255 (8 waves)

    if (i == 0) {                            // keep[0] = False: row excluded
        if (t == 0) { part_sum[0] = 0.0f; part_cnt[0] = 0u; }
        return;
    }

    for (int j = t; j < N_EMB; j += 256) {
        s_pd[j]  = pd[i * N_EMB + j];
        s_lab[j] = ind[j];
    }
    __syncthreads();

    const int labi = s_lab[i];
    if (t == 0) {                            // deterministic positive list
        int np = 0;
        for (int j = 0; j < N_EMB; ++j)
            if (s_lab[j] == labi) s_pos[np++] = j;
        s_npos = np;
    }
    __syncthreads();

    const int np = s_npos;
    float    lsum = 0.0f;
    unsigned lcnt = 0u;
    for (int k = t; k < N_EMB; k += 256) {
        if (s_lab[k] != labi) {              // negative
            const float nk = s_pd[k];
            for (int jj = 0; jj < np; ++jj) {
                float td = s_pd[s_pos[jj]] - nk + MARGIN;
                if (td > 0.0f) { lsum += td; lcnt++; }
            }
        }
    }

    #pragma unroll
    for (int off = 16; off > 0; off >>= 1) {
        lsum += __shfl_down(lsum, off, 32);
        lcnt += __shfl_down(lcnt, off, 32);
    }
    const int wid = t >> 5, lid = t & 31;
    if (lid == 0) { red_s[wid] = lsum; red_c[wid] = lcnt; }
    __syncthreads();
    if (t == 0) {
        float    S = 0.0f; unsigned C = 0u;
        #pragma unroll
        for (int w = 0; w < 8; ++w) { S += red_s[w]; C += red_c[w]; }
        part_sum[i] = S;
        part_cnt[i] = C;
    }
}

// ---------------------------------------------------------------------------
__global__ void finalize_kernel(const float* __restrict__ part_sum,
                                const unsigned int* __restrict__ part_cnt,
                                float* __restrict__ out) {
    if (threadIdx.x == 0) {
        float S = 0.0f;
        unsigned long long C = 0ull;
        for (int i = 0; i < N_EMB; ++i) { S += part_sum[i]; C += part_cnt[i]; }
        float c = (float)C;
        out[0] = (C > 0ull) ? S / fmaxf(c, 1.0f) : 0.0f;
    }
}

// ---------------------------------------------------------------------------
extern "C" void kernel_launch(void* const* d_in, const int* in_sizes, int n_in,
                              void* d_out, int out_size, void* d_ws, size_t ws_size,
                              hipStream_t stream) {
    const float* anchor   = (const float*)d_in[0];
    const float* positive = (const float*)d_in[1];
    const float* negative = (const float*)d_in[2];
    const int*   ind      = (const int*)d_in[3];
    float* out = (float*)d_out;

    float* ws   = (float*)d_ws;
    float* emb  = ws + WS_EMB;
    float* sq   = ws + WS_SQ;
    float* psum = ws + WS_SUM;
    unsigned int* pcnt = (unsigned int*)(ws + WS_CNT);
    float* pd   = ws + WS_PD;

    (void)in_sizes; (void)n_in; (void)out_size; (void)ws_size;

    // 1) emb = concat(anchor, positive, negative)
    concat_kernel<<<(N_EMB * DIM + 255) / 256, 256, 0, stream>>>(
        anchor, positive, negative, emb);

    // 2) squared norms (one wave per row)
    sq_kernel<<<N_EMB, 32, 0, stream>>>(emb, sq);

    // 3) pd tiles via f32 WMMA (one wave per 16x16 tile)
    dim3 grid(TILES, TILES);
    pd_wmma_kernel<<<grid, 32, 0, stream>>>(emb, sq, pd);

    // 4) per-row triplet reduction (factored mask, deterministic)
    triplet_reduce_kernel<<<N_EMB, 256, 0, stream>>>(pd, ind, psum, pcnt);

    // 5) finalize
    finalize_kernel<<<1, 32, 0, stream>>>(psum, pcnt, out);
}